// GCN_31284541784605
// MI455X (gfx1250) — compile-verified
//
#include <hip/hip_runtime.h>

#define N_NODES 50000
#define N_EDGES 800000
#define DIM     128

typedef float v2f __attribute__((ext_vector_type(2)));
typedef float v8f __attribute__((ext_vector_type(8)));

// ---------------------------------------------------------------------------
// Degree / normalization
// ---------------------------------------------------------------------------
__global__ __launch_bounds__(256) void k_init_deg(float* deg) {
    int i = blockIdx.x * 256 + threadIdx.x;
    if (i < N_NODES) deg[i] = 1.0f;                 // self-loop
}

__global__ __launch_bounds__(256) void k_count_deg(const int* __restrict__ dst, float* deg) {
    int e = blockIdx.x * 256 + threadIdx.x;         // E == 3125*256 exactly
    unsafeAtomicAdd(&deg[dst[e]], 1.0f);
}

__global__ __launch_bounds__(256) void k_finalize_dinv(float* dinv) {
    int i = blockIdx.x * 256 + threadIdx.x;
    if (i < N_NODES) dinv[i] = rsqrtf(dinv[i]);     // deg >= 1 always
}

// ---------------------------------------------------------------------------
// Dense GEMM: out[N,128] = A[N,128] @ W[128,128], exact f32 via WMMA 16x16x4.
// One wave per 16x16 output tile; 8 waves/block = the 8 column tiles;
// blockIdx.x = row tile (N/16 = 3125 exact).
//
// A-frag (16x4 f32, ISA 7.12.2): lanes 0-15 hold K=k,k+1; lanes 16-31 K=k+2,k+3.
// B-frag mirrors A with lanes indexing N. C/D: VGPR r -> M = r + 8*(lane>=16).
// ---------------------------------------------------------------------------
__global__ __launch_bounds__(256) void k_gemm_wmma(const float* __restrict__ A,
                                                   const float* __restrict__ W,
                                                   float* __restrict__ out) {
    const int lane = threadIdx.x & 31;
    const int wave = threadIdx.x >> 5;          // column tile 0..7
    const int row0 = blockIdx.x * 16;
    const int n0   = wave * 16;
    const int h    = lane >> 4;                 // half-wave select
    const int l    = lane & 15;

    const float* arow = A + (size_t)(row0 + l) * DIM;
    v8f c = {0.f, 0.f, 0.f, 0.f, 0.f, 0.f, 0.f, 0.f};

#pragma unroll
    for (int k = 0; k < DIM; k += 4) {
        const int ka = k + 2 * h;               // even -> 8B aligned pair load
        v2f a = *(const v2f*)(arow + ka);       // A[l][ka], A[l][ka+1]
        v2f b;
        b.x = W[(size_t)ka * DIM + n0 + l];     // W[ka][n]
        b.y = W[(size_t)(ka + 1) * DIM + n0 + l];
        c = __builtin_amdgcn_wmma_f32_16x16x4_f32(
                /*neg_a=*/false, a, /*neg_b=*/false, b,
                /*c_mod=*/(short)0, c, /*reuse_a=*/false, /*reuse_b=*/false);
    }

#pragma unroll
    for (int r = 0; r < 8; ++r)
        out[(size_t)(row0 + r + 8 * h) * DIM + n0 + l] = c[r];
}

// ---------------------------------------------------------------------------
// out[i] = t[i] * dinv[node]^2   (self-loop contribution, also zero-fills out)
// ---------------------------------------------------------------------------
__global__ __launch_bounds__(256) void k_selfloop_init(const float* __restrict__ t,
                                                       const float* __restrict__ dinv,
                                                       float* __restrict__ out) {
    int i = blockIdx.x * 256 + threadIdx.x;     // N*DIM == 25000*256 exactly
    float di = dinv[i >> 7];
    out[i] = t[i] * di * di;
}

// ---------------------------------------------------------------------------
// Edge scatter: one wave per edge; lane handles 4 channels.
// out[dst] += t[src] * dinv[src]*dinv[dst]
// ---------------------------------------------------------------------------
__global__ __launch_bounds__(256) void k_edge_scatter(const float* __restrict__ t,
                                                      float* __restrict__ out,
                                                      const int* __restrict__ src,
                                                      const int* __restrict__ dst,
                                                      const float* __restrict__ dinv) {
    const int lane = threadIdx.x & 31;
    const int wave = threadIdx.x >> 5;
    const int e = blockIdx.x * 8 + wave;        // E == 100000*8 exactly
    const int s = src[e];
    const int d = dst[e];
    const float norm = dinv[s] * dinv[d];
    const float4 v = ((const float4*)(t + (size_t)s * DIM))[lane];
    float* o = out + (size_t)d * DIM + lane * 4;
    unsafeAtomicAdd(o + 0, v.x * norm);
    unsafeAtomicAdd(o + 1, v.y * norm);
    unsafeAtomicAdd(o + 2, v.z * norm);
    unsafeAtomicAdd(o + 3, v.w * norm);
}

// ---------------------------------------------------------------------------
// h[i] = relu(h[i] + b[i % 128])
// ---------------------------------------------------------------------------
__global__ __launch_bounds__(256) void k_bias_relu(float* __restrict__ h,
                                                   const float* __restrict__ b) {
    int i = blockIdx.x * 256 + threadIdx.x;
    float v = h[i] + b[i & (DIM - 1)];
    h[i] = v > 0.f ? v : 0.f;
}

// ---------------------------------------------------------------------------
// Column-sum reduction (for mean pool)
// ---------------------------------------------------------------------------
__global__ void k_zero_g(float* g) { g[threadIdx.x] = 0.f; }

__global__ __launch_bounds__(256) void k_colsum(const float* __restrict__ h,
                                                float* __restrict__ gsum) {
    const int c  = threadIdx.x & (DIM - 1);
    const int r0 = blockIdx.x * 2 + (threadIdx.x >> 7);
    const int rs = gridDim.x * 2;
    float acc = 0.f;
    for (int r = r0; r < N_NODES; r += rs)
        acc += h[(size_t)r * DIM + c];
    unsafeAtomicAdd(&gsum[c], acc);
}

// ---------------------------------------------------------------------------
// out[j] = bl[j] + sum_k (gsum[k]/N) * Wl[k][j]    (one block of 128)
// ---------------------------------------------------------------------------
__global__ __launch_bounds__(128) void k_final_linear(const float* __restrict__ gsum,
                                                      const float* __restrict__ Wl,
                                                      const float* __restrict__ bl,
                                                      float* __restrict__ out) {
    const int j = threadIdx.x;
    const float invn = 1.0f / (float)N_NODES;
    float acc = bl[j];
#pragma unroll 8
    for (int k = 0; k < DIM; ++k)
        acc += gsum[k] * invn * Wl[(size_t)k * DIM + j];
    out[j] = acc;
}

// ---------------------------------------------------------------------------
extern "C" void kernel_launch(void* const* d_in, const int* in_sizes, int n_in,
                              void* d_out, int out_size, void* d_ws, size_t ws_size,
                              hipStream_t stream) {
    const float* x  = (const float*)d_in[0];
    const int*   ei = (const int*)d_in[1];      // [2, E] flat
    const float* W1 = (const float*)d_in[2];
    const float* b1 = (const float*)d_in[3];
    const float* W2 = (const float*)d_in[4];
    const float* b2 = (const float*)d_in[5];
    const float* Wl = (const float*)d_in[6];
    const float* bl = (const float*)d_in[7];
    float* out = (float*)d_out;

    const int* src = ei;
    const int* dst = ei + N_EDGES;

    float* bufA = (float*)d_ws;                         // GEMM output t
    float* bufB = bufA + (size_t)N_NODES * DIM;         // aggregation / h
    float* dinv = bufB + (size_t)N_NODES * DIM;         // degree -> dinv
    float* gsum = dinv + N_NODES;                       // 128 column sums

    const int nodeBlks = (N_NODES + 255) / 256;         // 196
    const int ndBlks   = (N_NODES * DIM) / 256;         // 25000
    const int rowTiles = N_NODES / 16;                  // 3125

    // normalization
    k_init_deg<<<nodeBlks, 256, 0, stream>>>(dinv);
    k_count_deg<<<N_EDGES / 256, 256, 0, stream>>>(dst, dinv);
    k_finalize_dinv<<<nodeBlks, 256, 0, stream>>>(dinv);

    // ---- layer 1: h1 = relu(aggregate(x @ W1) + b1) -> bufB
    k_gemm_wmma<<<rowTiles, 256, 0, stream>>>(x, W1, bufA);
    k_selfloop_init<<<ndBlks, 256, 0, stream>>>(bufA, dinv, bufB);
    k_edge_scatter<<<N_EDGES / 8, 256, 0, stream>>>(bufA, bufB, src, dst, dinv);
    k_bias_relu<<<ndBlks, 256, 0, stream>>>(bufB, b1);

    // ---- layer 2: h2 = relu(aggregate(h1 @ W2) + b2) -> bufB (ping-pong)
    k_gemm_wmma<<<rowTiles, 256, 0, stream>>>(bufB, W2, bufA);
    k_selfloop_init<<<ndBlks, 256, 0, stream>>>(bufA, dinv, bufB);
    k_edge_scatter<<<N_EDGES / 8, 256, 0, stream>>>(bufA, bufB, src, dst, dinv);
    k_bias_relu<<<ndBlks, 256, 0, stream>>>(bufB, b2);

    // ---- mean pool + final linear
    k_zero_g<<<1, DIM, 0, stream>>>(gsum);
    k_colsum<<<256, 256, 0, stream>>>(bufB, gsum);
    k_final_linear<<<1, DIM, 0, stream>>>(gsum, Wl, bl, out);
}